// ReverseKlLoss_21036749815963
// MI455X (gfx1250) — compile-verified
//
#include <hip/hip_runtime.h>

// -----------------------------------------------------------------------------
// Reverse-KL loss, MI455X (gfx1250, wave32).
//   loss = sum_ij p_ij * (log(p_ij + eps) - log(q_ij + eps)),  p = softmax(out)
// Memory-bound: 160 MB single-pass stream -> ~6.9 us floor at 23.3 TB/s.
// Kernel 1: grid-stride b128 streaming + per-block deterministic reduction.
// Kernel 2: single wave32; final 32->1 reduction via V_WMMA_F32_16X16X4_F32
//           (A = ones, B = lane partials -> D row 0 holds column sums).
// -----------------------------------------------------------------------------

typedef __attribute__((ext_vector_type(2))) float v2f;
typedef __attribute__((ext_vector_type(8))) float v8f;

#define NBLK 2048
#define TPB  256

// log(q + 1e-10) for q in {0.9, 0.1, 0.0}
#define LQ9 (-0.10536051554f)   // log(0.9 + 1e-10)
#define LQ1 (-2.30258508296f)   // log(0.1 + 1e-10)
#define LQE (-23.02585092994f)  // log(1e-10)

__device__ __forceinline__ float row_term(float4 x, float t) {
    // target row selection via cndmask chains (no indexed table loads)
    const bool is0 = (t == 0.0f);
    const bool is1 = (t == 1.0f);
    const bool is2 = (t == 2.0f);
    const float lq0 = is0 ? LQ9 : (is1 ? LQ1 : LQE);
    const float lq1 = is0 ? LQ1 : (is1 ? LQ9 : LQE);
    const float lq2 = is2 ? LQ9 : ((is0 || is1) ? LQE : LQ1);
    const float lq3 = is2 ? LQ1 : ((is0 || is1) ? LQE : LQ9);

    // stable softmax + log-sum-exp:  log p_i = z_i - log S
    const float m  = fmaxf(fmaxf(x.x, x.y), fmaxf(x.z, x.w));
    const float z0 = x.x - m, z1 = x.y - m, z2 = x.z - m, z3 = x.w - m;
    const float e0 = __expf(z0), e1 = __expf(z1), e2 = __expf(z2), e3 = __expf(z3);
    const float S   = (e0 + e1) + (e2 + e3);
    const float inv = __builtin_amdgcn_rcpf(S);   // v_rcp_f32
    const float lnS = __logf(S);                  // v_log_f32

    return e0 * inv * (z0 - lnS - lq0)
         + e1 * inv * (z1 - lnS - lq1)
         + e2 * inv * (z2 - lnS - lq2)
         + e3 * inv * (z3 - lnS - lq3);
}

__global__ __launch_bounds__(TPB) void rkl_partial_kernel(
    const float* __restrict__ logits,   // [nrows, 4]
    const float* __restrict__ target,   // [nrows]
    float* __restrict__ partials,       // [NBLK]
    int nrows)
{
    const int tid    = blockIdx.x * TPB + threadIdx.x;
    const int stride = NBLK * TPB;
    const int nchunk = (nrows + 3) >> 2;   // 4 rows per chunk

    float acc = 0.0f;
    for (int c = tid; c < nchunk; c += stride) {
        const int r0 = c << 2;
        if (r0 + 3 < nrows) {
            // 4 targets in one b128, 4 rows of logits as 4x b128 (64B contiguous)
            const float4 tg = *(const float4*)(target + r0);
            const float4 x0 = *(const float4*)(logits + (size_t)(r0 + 0) * 4);
            const float4 x1 = *(const float4*)(logits + (size_t)(r0 + 1) * 4);
            const float4 x2 = *(const float4*)(logits + (size_t)(r0 + 2) * 4);
            const float4 x3 = *(const float4*)(logits + (size_t)(r0 + 3) * 4);
            acc += row_term(x0, tg.x);
            acc += row_term(x1, tg.y);
            acc += row_term(x2, tg.z);
            acc += row_term(x3, tg.w);
        } else {
            for (int r = r0; r < nrows; ++r) {
                const float4 x = *(const float4*)(logits + (size_t)r * 4);
                acc += row_term(x, target[r]);
            }
        }
    }

    // wave32 xor-shuffle reduction
    acc += __shfl_xor(acc, 16, 32);
    acc += __shfl_xor(acc,  8, 32);
    acc += __shfl_xor(acc,  4, 32);
    acc += __shfl_xor(acc,  2, 32);
    acc += __shfl_xor(acc,  1, 32);

    __shared__ float wsum[TPB / 32];
    const int lane = threadIdx.x & 31;
    const int wid  = threadIdx.x >> 5;
    if (lane == 0) wsum[wid] = acc;
    __syncthreads();

    if (threadIdx.x < 32) {
        float v = (threadIdx.x < (TPB / 32)) ? wsum[threadIdx.x] : 0.0f;
        v += __shfl_xor(v, 4, 32);
        v += __shfl_xor(v, 2, 32);
        v += __shfl_xor(v, 1, 32);
        if (threadIdx.x == 0) partials[blockIdx.x] = v;
    }
}

// Single wave32. Final 32-lane reduction via V_WMMA_F32_16X16X4_F32:
// A = 16x4 all-ones, B carries the 32 partials (one (k,n) slot each, rest 0)
// => D[m][n] = colsum_n(B)  =>  sum_{n=0..15} D[0][n] = exact f32 total.
__global__ __launch_bounds__(32) void rkl_final_kernel(
    const float* __restrict__ partials,
    float* __restrict__ out,
    int nblk)
{
    const int lane = threadIdx.x;
    float acc = 0.0f;
    for (int i = lane; i < nblk; i += 32) acc += partials[i];

    v2f a; a[0] = 1.0f; a[1] = 1.0f;   // A[m][k] = 1 for all 64 slots
    v2f b; b[0] = acc;  b[1] = 0.0f;   // 32 values somewhere in B, zeros elsewhere
    v8f c = {};
    // (neg_a, A, neg_b, B, c_mod, C, reuse_a, reuse_b) — EXEC is all-1s here
    v8f d = __builtin_amdgcn_wmma_f32_16x16x4_f32(
        false, a, false, b, (short)0, c, false, false);

    // D[0][n] lives in d[0] on lanes n = 0..15; sum them (xor stays in-half)
    float t = d[0];
    t += __shfl_xor(t, 8, 32);
    t += __shfl_xor(t, 4, 32);
    t += __shfl_xor(t, 2, 32);
    t += __shfl_xor(t, 1, 32);
    if (lane == 0) out[0] = t;
}

extern "C" void kernel_launch(void* const* d_in, const int* in_sizes, int n_in,
                              void* d_out, int out_size, void* d_ws, size_t ws_size,
                              hipStream_t stream) {
    const float* logits = (const float*)d_in[0];   // [N, 4] f32
    const float* target = (const float*)d_in[1];   // [N]    f32
    const int nrows = in_sizes[1];

    float* partials = (float*)d_ws;                // NBLK floats (8 KB)

    rkl_partial_kernel<<<NBLK, TPB, 0, stream>>>(logits, target, partials, nrows);
    rkl_final_kernel<<<1, 32, 0, stream>>>(partials, (float*)d_out, NBLK);
}